// LocalFeatureLoss_53661321396474
// MI455X (gfx1250) — compile-verified
//
#include <hip/hip_runtime.h>

// LocalFeatureLoss training path for MI455X (gfx1250, wave32, WMMA).
// B=32, L=H*W=1024, C=256.  M[a,j] = <p[a], q[j]> per batch.
// fp32 GEMM on V_WMMA_F32_16X16X4_F32 with 2x2 register blocking:
// 4 b64 loads feed 4 WMMAs (1 VMEM per WMMA), M never materialized.

#define BATCH 32
#define L     1024
#define CDIM  256
#define NRB   4        // row blocks per batch (256 rows each)
#define NEG_INF (-3.402823466e38f)

typedef __attribute__((ext_vector_type(2))) float v2f;
typedef __attribute__((ext_vector_type(8))) float v8f;

// ---------------------------------------------------------------------------
// Kernel 1: grid = BATCH*NRB blocks, 256 threads (8 waves).
// Wave w owns rows [a0, a0+32) (two 16-row A frags) and sweeps all 1024
// columns in 32-wide tiles (two 16-col B frags) -> 2x2 accumulators.
// Produces: max2 (row argmax) for its rows, per-block column max (val+row).
// ---------------------------------------------------------------------------
__global__ __launch_bounds__(256)
void lfl_gemm_argmax_kernel(const float* __restrict__ p,   // feats2 (preds)
                            const float* __restrict__ q,   // feats1 (queries)
                            float* __restrict__ colvalg,   // [B][NRB][L]
                            int*   __restrict__ colidxg,   // [B][NRB][L]
                            int*   __restrict__ max2g)     // [B][L]
{
    __shared__ float s_colv[8 * L];
    __shared__ int   s_coli[8 * L];

    const int b    = blockIdx.x >> 2;
    const int rb   = blockIdx.x & 3;
    const int w    = threadIdx.x >> 5;
    const int lane = threadIdx.x & 31;
    const int hi   = lane >> 4;     // wave half
    const int m    = lane & 15;     // row-in-frag for A, col-in-frag for B/D

    const int a0 = rb * 256 + w * 32;   // this wave's 32-row strip of P

    const float* __restrict__ Pb = p + (size_t)b * L * CDIM;
    const float* __restrict__ Qb = q + (size_t)b * L * CDIM;
    // A frag element (row, K): lane (hi,m), VGPR v holds P[row0+m][c0+2*hi+v]
    const float* __restrict__ Arow0 = Pb + (size_t)(a0 + m) * CDIM + 2 * hi;
    const float* __restrict__ Arow1 = Arow0 + 16 * CDIM;

    // running row-max: acc[rt][ct] VGPR v holds row a0 + 16*rt + 8*hi + v
    float rv[16];
    int   ri[16];
#pragma unroll
    for (int i = 0; i < 16; ++i) { rv[i] = NEG_INF; ri[i] = 0; }

    for (int j0 = 0; j0 < L; j0 += 32) {
        const float* __restrict__ Brow0 = Qb + (size_t)(j0 + m) * CDIM + 2 * hi;
        const float* __restrict__ Brow1 = Brow0 + 16 * CDIM;

        v8f acc[2][2];
#pragma unroll
        for (int rt = 0; rt < 2; ++rt)
#pragma unroll
            for (int ct = 0; ct < 2; ++ct)
                acc[rt][ct] = (v8f){0.f, 0.f, 0.f, 0.f, 0.f, 0.f, 0.f, 0.f};

#pragma unroll 4
        for (int c0 = 0; c0 < CDIM; c0 += 4) {
            v2f aA = *(const v2f*)(Arow0 + c0);
            v2f aB = *(const v2f*)(Arow1 + c0);
            v2f bA = *(const v2f*)(Brow0 + c0);
            v2f bB = *(const v2f*)(Brow1 + c0);
            // (neg_a, A, neg_b, B, c_mod, C, reuse_a, reuse_b)
            acc[0][0] = __builtin_amdgcn_wmma_f32_16x16x4_f32(
                false, aA, false, bA, (short)0, acc[0][0], false, false);
            acc[0][1] = __builtin_amdgcn_wmma_f32_16x16x4_f32(
                false, aA, false, bB, (short)0, acc[0][1], false, false);
            acc[1][0] = __builtin_amdgcn_wmma_f32_16x16x4_f32(
                false, aB, false, bA, (short)0, acc[1][0], false, false);
            acc[1][1] = __builtin_amdgcn_wmma_f32_16x16x4_f32(
                false, aB, false, bB, (short)0, acc[1][1], false, false);
        }

        // ---- per column-frag reductions (branchless selects) ----
#pragma unroll
        for (int ct = 0; ct < 2; ++ct) {
            const int jcol = j0 + 16 * ct + m;

            // column max over this wave's 32 rows: in-thread (ascending rows),
            // then exchange wave halves (row order interleaves -> index tiebreak)
            float cv = acc[0][ct][0];
            int   cr = a0 + 8 * hi;
#pragma unroll
            for (int rt = 0; rt < 2; ++rt)
#pragma unroll
                for (int v = 0; v < 8; ++v) {
                    if (rt == 0 && v == 0) continue;
                    float val = acc[rt][ct][v];
                    int   row = a0 + 16 * rt + 8 * hi + v;
                    bool  t   = val > cv;          // strict '>' keeps first occurrence
                    cv = t ? val : cv;
                    cr = t ? row : cr;
                }
            {
                float ov = __shfl_xor(cv, 16);
                int   oi = __shfl_xor(cr, 16);
                bool  t  = (ov > cv) || (ov == cv && oi < cr);
                cv = t ? ov : cv;
                cr = t ? oi : cr;
            }
            if (hi == 0) {
                s_colv[w * L + jcol] = cv;
                s_coli[w * L + jcol] = cr;
            }

            // running row max (ascending column order: ct=0 then ct=1)
#pragma unroll
            for (int rt = 0; rt < 2; ++rt)
#pragma unroll
                for (int v = 0; v < 8; ++v) {
                    float val = acc[rt][ct][v];
                    int   k   = rt * 8 + v;
                    bool  u   = val > rv[k];
                    rv[k] = u ? val : rv[k];
                    ri[k] = u ? jcol : ri[k];
                }
        }
    }

    // reduce row max across the 16 lanes of each half (masks 1..8 stay in-half)
#pragma unroll
    for (int mask = 1; mask <= 8; mask <<= 1) {
#pragma unroll
        for (int k = 0; k < 16; ++k) {
            float ov = __shfl_xor(rv[k], mask);
            int   oi = __shfl_xor(ri[k], mask);
            bool  t  = (ov > rv[k]) || (ov == rv[k] && oi < ri[k]);
            rv[k] = t ? ov : rv[k];
            ri[k] = t ? oi : ri[k];
        }
    }
    if (m == 0) {
#pragma unroll
        for (int rt = 0; rt < 2; ++rt)
#pragma unroll
            for (int v = 0; v < 8; ++v)
                max2g[b * L + a0 + 16 * rt + 8 * hi + v] = ri[rt * 8 + v];
    }

    __syncthreads();

    // fold the 8 per-wave column maxes (ascending wave = ascending rows)
    for (int j = threadIdx.x; j < L; j += 256) {
        float bv = s_colv[j];
        int   bi = s_coli[j];
#pragma unroll
        for (int ww = 1; ww < 8; ++ww) {
            float vv = s_colv[ww * L + j];
            bool  t  = vv > bv;               // ascending rows: keep first on tie
            bi = t ? s_coli[ww * L + j] : bi;
            bv = t ? vv : bv;
        }
        colvalg[(size_t)(b * NRB + rb) * L + j] = bv;
        colidxg[(size_t)(b * NRB + rb) * L + j] = bi;
    }
}

// ---------------------------------------------------------------------------
// Kernel 2: per batch, fold the NRB row-block column maxes -> (sims, max1),
// mutual-NN check against max2, deterministic masked-mean reduction.
// grid = BATCH blocks, 256 threads.
// ---------------------------------------------------------------------------
__global__ __launch_bounds__(256)
void lfl_finalize_kernel(const float* __restrict__ colvalg,
                         const int*   __restrict__ colidxg,
                         const int*   __restrict__ max2g,
                         float*       __restrict__ out)
{
    const int b   = blockIdx.x;
    const int tid = threadIdx.x;

    float sum = 0.f;
    int   cnt = 0;
    for (int j = tid; j < L; j += 256) {
        float bv = colvalg[(size_t)(b * NRB) * L + j];
        int   bi = colidxg[(size_t)(b * NRB) * L + j];
#pragma unroll
        for (int rb = 1; rb < NRB; ++rb) {
            float vv = colvalg[(size_t)(b * NRB + rb) * L + j];
            bool  t  = vv > bv;               // ascending row blocks
            bi = t ? colidxg[(size_t)(b * NRB + rb) * L + j] : bi;
            bv = t ? vv : bv;
        }
        // mutual nearest-neighbor: max2[max1[j]] == j
        int mj = max2g[b * L + bi];
        bool valid = (mj == j);
        sum += valid ? bv : 0.f;
        cnt += valid ? 1 : 0;
    }

    __shared__ float sv[256];
    __shared__ int   sc[256];
    sv[tid] = sum;
    sc[tid] = cnt;
    __syncthreads();
    for (int s = 128; s > 0; s >>= 1) {
        if (tid < s) { sv[tid] += sv[tid + s]; sc[tid] += sc[tid + s]; }
        __syncthreads();
    }
    if (tid == 0)
        out[b] = sv[0] / fmaxf((float)sc[0], 1.0f);
}

// ---------------------------------------------------------------------------
extern "C" void kernel_launch(void* const* d_in, const int* in_sizes, int n_in,
                              void* d_out, int out_size, void* d_ws, size_t ws_size,
                              hipStream_t stream) {
    const float* q = (const float*)d_in[0];  // feats1 -> queries
    const float* p = (const float*)d_in[1];  // feats2 -> preds
    float* out = (float*)d_out;              // (B,) fp32

    // workspace layout
    float* colvalg = (float*)d_ws;                                         // B*NRB*L f32
    int*   colidxg = (int*)((char*)d_ws + (size_t)BATCH * NRB * L * 4);    // B*NRB*L i32
    int*   max2g   = (int*)((char*)d_ws + (size_t)BATCH * NRB * L * 8);    // B*L     i32

    lfl_gemm_argmax_kernel<<<BATCH * NRB, 256, 0, stream>>>(p, q, colvalg, colidxg, max2g);
    lfl_finalize_kernel<<<BATCH, 256, 0, stream>>>(colvalg, colidxg, max2g, out);
}